// DeEmph_50603304682089
// MI455X (gfx1250) — compile-verified
//
#include <hip/hip_runtime.h>

// De-emphasis IIR scan: x[n] = C*x[n-1] + e[n], per row, 64 rows x 480000 f32.
//
// Memory-bound (246 MB min traffic ~10.5us @ 23.3 TB/s). Three passes:
//  pass1: read input once (lands in 192MB L2; 123MB input fits), per-segment
//         scan aggregates.
//  pass2: tiny per-row serial scan of segment aggregates -> segment carries.
//  pass3: re-read input (L2 hit), full scan via WMMA scan-as-matmul, write
//         output with non-temporal stores so output doesn't evict L2 input.
//
// Pass 3 does the 256-sample-tile scan as two chained f32 WMMA matmuls
// (V_WMMA_F32_16X16X4_F32, K=4 slices x4):
//   Y = E' x T           (T[k][n] = C^(n-k), k<=n : intra-16 scan)
//   X = A2 x B2 + Y      (A2[m][p] = (C^16)^(m-1-p), p<m ;
//                         B2[p][n] = Y[p][15] * C^(n+1) : cross-16 carry)
// Incoming tile carry c is folded as E'[0][0] = E[0][0] + C*c (single FMA,
// lane-0-masked), so X is directly the final scan; carry_out = X[15][15].
//
// All C^k powers are built with raw v_exp_f32 (exp2(k*log2C)) instead of
// ocml powf: 3 VALU per power, no overflow/class checks.

#define COEFF 0.95f
#define LOG2_COEFF (-0.07400058144377693f)   // log2(0.95)

static constexpr int ROWS         = 64;
static constexpr int NCOLS        = 480000;
static constexpr int TILE         = 256;     // 16x16 matrix tile
static constexpr int TILES_PER_SEG= 15;
static constexpr int SEG          = TILE * TILES_PER_SEG;   // 3840
static constexpr int SEGS_PER_ROW = NCOLS / SEG;            // 125
static constexpr int TOTAL_WAVES  = ROWS * SEGS_PER_ROW;    // 8000
static constexpr int WAVES_PER_BLK= 8;                       // 256 thr
static constexpr int GRID         = TOTAL_WAVES / WAVES_PER_BLK; // 1000

typedef __attribute__((ext_vector_type(2))) float v2f;
typedef __attribute__((ext_vector_type(8))) float v8f;

static __device__ __forceinline__ float readlane_f(float v, int lane) {
    int i = __builtin_amdgcn_readlane(__builtin_bit_cast(int, v), lane);
    return __builtin_bit_cast(float, i);
}

// C^k via v_exp_f32; k up to ~3840 -> arg down to -284 -> flushes to 0 (exact)
static __device__ __forceinline__ float cpow(int k) {
#if defined(__AMDGCN__) && __has_builtin(__builtin_amdgcn_exp2f)
    return __builtin_amdgcn_exp2f((float)k * LOG2_COEFF);
#else
    return exp2f((float)k * LOG2_COEFF);
#endif
}

// ---------------- Pass 1: per-segment aggregate  B_seg = sum c^(L-1-j) e[j]
__global__ __launch_bounds__(256) void deemph_pass1(
        const float* __restrict__ in, float* __restrict__ agg) {
    const int lane = threadIdx.x & 31;
    const int hi   = lane >> 4;        // half-wave select
    const int m    = lane & 15;        // matrix row this lane owns (A layout)
    const int wave = blockIdx.x * WAVES_PER_BLK + (threadIdx.x >> 5);
    const int row  = wave / SEGS_PER_ROW;
    const int seg  = wave % SEGS_PER_ROW;
    const size_t base = (size_t)row * NCOLS + (size_t)seg * SEG;

    // per-lane weights: lane touches elements idx = 16*m + 4q + 2*hi + j
    float w[4][2];
#pragma unroll
    for (int q = 0; q < 4; ++q)
#pragma unroll
        for (int j = 0; j < 2; ++j) {
            int idx = 16 * m + 4 * q + 2 * hi + j;
            w[q][j] = cpow(255 - idx);
        }
    const float c256 = cpow(256);

    // single base pointer; all tile/slice offsets are immediates
    const float* tp0 = in + base + 16 * m + 2 * hi;

    float acc = 0.0f;
#pragma unroll
    for (int t = 0; t < TILES_PER_SEG; ++t) {
        float partial = 0.0f;
#pragma unroll
        for (int q = 0; q < 4; ++q) {
            v2f v = *reinterpret_cast<const v2f*>(tp0 + t * TILE + 4 * q);
            partial = fmaf(w[q][0], v[0], partial);
            partial = fmaf(w[q][1], v[1], partial);
        }
        acc = fmaf(acc, c256, partial);   // shift previous tiles by c^256
    }
    // cross-lane sum (wave32)
#pragma unroll
    for (int off = 16; off >= 1; off >>= 1)
        acc += __shfl_xor(acc, off, 32);
    if (lane == 0) agg[wave] = acc;
}

// ---------------- Pass 2: scan segment aggregates -> incoming carries
__global__ void deemph_pass2(const float* __restrict__ agg,
                             float* __restrict__ carry) {
    const int row = threadIdx.x;        // 64 threads
    const float cL = cpow(SEG);         // exp2(-284) -> 0 (correct limit)
    float c = 0.0f;
    for (int s = 0; s < SEGS_PER_ROW; ++s) {
        carry[row * SEGS_PER_ROW + s] = c;
        c = fmaf(c, cL, agg[row * SEGS_PER_ROW + s]);
    }
}

// ---------------- Pass 3: full scan via WMMA scan-as-matmul
__global__ __launch_bounds__(256) void deemph_pass3(
        const float* __restrict__ in, const float* __restrict__ carries,
        float* __restrict__ out) {
    const int lane = threadIdx.x & 31;
    const int hi   = lane >> 4;
    const int mn   = lane & 15;     // A-operand: row M; B/C/D-operand: col N
    const int wave = blockIdx.x * WAVES_PER_BLK + (threadIdx.x >> 5);
    const int row  = wave / SEGS_PER_ROW;
    const int seg  = wave % SEGS_PER_ROW;
    const size_t base = (size_t)row * NCOLS + (size_t)seg * SEG;

    // Constant operands, built in documented gfx1250 WMMA layouts.
    // B-operand 4x16 slices of T: v_j holds row k = 4q + j + 2*hi.
    // A-operand 16x4 slices of A2: lane row = mn, v_j holds col p = 4q+j+2*hi.
    v2f  Tb[4], A2[4];
#pragma unroll
    for (int q = 0; q < 4; ++q)
#pragma unroll
        for (int j = 0; j < 2; ++j) {
            int k = 4 * q + j + 2 * hi;
            Tb[q][j] = (mn >= k) ? cpow(mn - k) : 0.0f;
            A2[q][j] = (k <  mn) ? cpow(16 * (mn - 1 - k)) : 0.0f;
        }
    const float cpn1 = cpow(mn + 1);      // C^(n+1)
    const float cpA  = hi ? 0.0f : cpn1;  // low half-wave takes rows 4q+j
    const float cpB  = hi ? cpn1 : 0.0f;  // high half-wave takes rows 4q+j+2
    const float sel0 = (lane == 0) ? COEFF : 0.0f;  // carry-inject mask

    float carry = carries[row * SEGS_PER_ROW + seg];

    // single base pointers; all per-tile offsets become immediates
    const float* tp0 = in  + base + 16 * mn + 2 * hi;
    float*       op0 = out + base + 128 * hi + mn;

#pragma unroll
    for (int t = 0; t < TILES_PER_SEG; ++t) {
        // Load E directly in A-operand layout (lane owns row mn; half-wave
        // picks K pair). Four b64 loads cover the contiguous 1KB tile.
        v2f Ea[4];
#pragma unroll
        for (int q = 0; q < 4; ++q)
            Ea[q] = *reinterpret_cast<const v2f*>(tp0 + t * TILE + 4 * q);

        // Fold incoming carry into the tile: E[0][0] += C*carry (lane 0 only).
        // A zero-init scan of the patched tile equals the carried scan.
        Ea[0][0] = fmaf(carry, sel0, Ea[0][0]);

        // Y = E' x T  (chain K=4 slices)
        v8f y = {0.f, 0.f, 0.f, 0.f, 0.f, 0.f, 0.f, 0.f};
#pragma unroll
        for (int q = 0; q < 4; ++q)
            y = __builtin_amdgcn_wmma_f32_16x16x4_f32(
                    false, Ea[q], false, Tb[q], (short)0, y, false, false);

        // Broadcast last column Y[p][15] (lanes 15 / 31 per C/D layout).
        // yl[15] feeds only the all-zero column 15 of A2 -> value irrelevant.
        float yl[16];
#pragma unroll
        for (int r = 0; r < 8; ++r) {
            yl[r] = readlane_f(y[r], 15);
            if (r < 7) yl[r + 8] = readlane_f(y[r], 31);
        }
        yl[15] = 0.0f;

        // X = A2 x B2 + Y ; B2 row (4q+j+2*hi) = Y[row][15] * C^(n+1),
        // built select-free with half-wave-masked coefficients.
        v8f x = y;
#pragma unroll
        for (int q = 0; q < 4; ++q) {
            v2f B2;
#pragma unroll
            for (int j = 0; j < 2; ++j) {
                const int p = 4 * q + j;
                B2[j] = fmaf(yl[p + 2], cpB, yl[p] * cpA);
            }
            x = __builtin_amdgcn_wmma_f32_16x16x4_f32(
                    false, A2[q], false, B2, (short)0, x, false, false);
        }

        // Outgoing carry = X[15][15] (VGPR 7, lane 31 in C/D layout).
        carry = readlane_f(x[7], 31);

        // Store in C/D layout; non-temporal so output doesn't evict the
        // L2-resident input other blocks still need.
#pragma unroll
        for (int r = 0; r < 8; ++r)
            __builtin_nontemporal_store(x[r], op0 + t * TILE + 16 * r);
    }
}

extern "C" void kernel_launch(void* const* d_in, const int* in_sizes, int n_in,
                              void* d_out, int out_size, void* d_ws, size_t ws_size,
                              hipStream_t stream) {
    (void)in_sizes; (void)n_in; (void)out_size; (void)ws_size;
    const float* in  = (const float*)d_in[0];
    float*       out = (float*)d_out;
    float*       agg = (float*)d_ws;              // 8000 floats
    float*       car = agg + TOTAL_WAVES;         // 8000 floats

    deemph_pass1<<<GRID, 256, 0, stream>>>(in, agg);
    deemph_pass2<<<1, ROWS, 0, stream>>>(agg, car);
    deemph_pass3<<<GRID, 256, 0, stream>>>(in, car, out);
}